// Decoder_luong_9964324127485
// MI455X (gfx1250) — compile-verified
//
#include <hip/hip_runtime.h>
#include <math.h>

// Shapes from the reference
#define BV 32000   // vocab
#define EE 1024    // embedding
#define HH 1024    // hidden
#define SS 1024    // seq len
#define BB 64      // batch

typedef __attribute__((ext_vector_type(2))) float v2f;
typedef __attribute__((ext_vector_type(8))) float v8f;
typedef __attribute__((ext_vector_type(4))) int   v4i;

#ifndef __has_builtin
#define __has_builtin(x) 0
#endif

#if __has_builtin(__builtin_amdgcn_global_load_async_to_lds_b128)
#define HAVE_ASYNC_LDS 1
typedef __attribute__((address_space(1))) v4i* gbl_b128_t;
typedef __attribute__((address_space(3))) v4i* lds_b128_t;
#else
#define HAVE_ASYNC_LDS 0
#endif

#if HAVE_ASYNC_LDS
#if __has_builtin(__builtin_amdgcn_s_wait_asynccnt)
#define WAIT_ASYNC(n) __builtin_amdgcn_s_wait_asynccnt(n)
#else
#define WAIT_ASYNC(n) asm volatile("s_wait_asynccnt %0" :: "i"(n) : "memory")
#endif
#else
#define WAIT_ASYNC(n) do {} while (0)
#endif

// 16-byte global -> LDS copy: async DMA (ASYNCcnt) on CDNA5, sync fallback.
__device__ __forceinline__ void copy16_to_lds(const float* g, float* l) {
#if HAVE_ASYNC_LDS
  __builtin_amdgcn_global_load_async_to_lds_b128(
      (gbl_b128_t)g, (lds_b128_t)l, /*offset=*/0, /*cpol=*/0);
#else
  *(float4*)l = *(const float4*)g;
#endif
}

// ---------------------------------------------------------------------------
// 1) x0 = concat(emb_table[word_input], last_context)  -> (B, E+H) = (64, 2048)
// ---------------------------------------------------------------------------
__global__ void build_x0_kernel(const int* __restrict__ word_input,
                                const float* __restrict__ last_context,
                                const float* __restrict__ emb_table,
                                float* __restrict__ x0) {
  int idx = blockIdx.x * blockDim.x + threadIdx.x;   // B * 2048
  int b = idx >> 11;
  int j = idx & 2047;
  float v;
  if (j < EE) {
    v = emb_table[(size_t)word_input[b] * EE + j];
  } else {
    v = last_context[b * HH + (j - EE)];
  }
  x0[idx] = v;
}

// ---------------------------------------------------------------------------
// 2) WMMA fp32 GEMM: C(64 x N) = A(64 x K) @ W(N x K)^T + bias
//    Block = 512 threads (16 waves) computing a 64x64 tile: wave w ->
//    (mi = w&3, ni = w>>2) 16x16 sub-tile.  K-blocked (KB=64) through
//    double-buffered LDS filled with async global->LDS DMA.
//    grid.x = N/64.
// ---------------------------------------------------------------------------
#define KB   64
#define LDSP 68   // padded row stride (floats): 272 B -> 16B aligned, conflict-free

__global__ void wmma_gemm_bias_kernel(const float* __restrict__ A,
                                      const float* __restrict__ W,
                                      const float* __restrict__ bias,
                                      float* __restrict__ C,
                                      int N, int K) {
  __shared__ float Alds[2][64 * LDSP];
  __shared__ float Wlds[2][64 * LDSP];

  const int tid   = threadIdx.x;
  const int wave  = tid >> 5;
  const int lane  = tid & 31;
  const int mi    = wave & 3;       // M-tile within 64 rows
  const int ni    = wave >> 2;      // N-tile within 64 cols
  const int row   = lane & 15;      // M row (A frag) / N col (B frag) within tile
  const int khalf = lane >> 4;      // 0: K=0,1   1: K=2,3  (ISA 16x4 fp32 layout)
  const int n0    = blockIdx.x * 64;

  const int lr = tid >> 4;          // 0..31  staging row
  const int lc = (tid & 15) * 4;    // staging col (float4)

  v8f acc = {};

  // Issue one K-block's staging: A 64x64 + W 64x64, 4 x b128 per thread.
  auto issue_tiles = [&](int kb, int buf) {
    #pragma unroll
    for (int p = 0; p < 2; ++p) {
      int rr = p * 32 + lr;
      copy16_to_lds(&A[(size_t)rr * K + kb + lc], &Alds[buf][rr * LDSP + lc]);
    }
    #pragma unroll
    for (int p = 0; p < 2; ++p) {
      int rr = p * 32 + lr;
      copy16_to_lds(&W[(size_t)(n0 + rr) * K + kb + lc],
                    &Wlds[buf][rr * LDSP + lc]);
    }
  };

  const int nk = K / KB;
  issue_tiles(0, 0);

  for (int i = 0; i < nk; ++i) {
    const int cur = i & 1;
    if (i + 1 < nk) {
      issue_tiles((i + 1) * KB, cur ^ 1);  // prefetch next block into other buf
      WAIT_ASYNC(4);                       // in-order: current buf's 4 are done
    } else {
      WAIT_ASYNC(0);
    }
    __syncthreads();                       // all waves' DMA landed

    const float* arow = &Alds[cur][(mi * 16 + row) * LDSP];
    const float* wrow = &Wlds[cur][(ni * 16 + row) * LDSP];
    #pragma unroll
    for (int kk = 0; kk < KB; kk += 4) {
      int o = kk + 2 * khalf;
      v2f a = { arow[o], arow[o + 1] };    // A[row][k], A[row][k+1]
      v2f b = { wrow[o], wrow[o + 1] };    // B[k][col]=W[col][k], B[k+1][col]
      acc = __builtin_amdgcn_wmma_f32_16x16x4_f32(
          /*neg_a=*/false, a, /*neg_b=*/false, b,
          /*c_mod=*/(short)0, acc, /*reuse_a=*/false, /*reuse_b=*/false);
    }
    __syncthreads();                       // done reading `cur` before reuse
  }

  // Store D: VGPR v -> row (v + 8*khalf), col = lane&15
  const int   col = n0 + ni * 16 + row;
  const float bv  = bias[col];
  #pragma unroll
  for (int v = 0; v < 8; ++v) {
    int m = mi * 16 + v + 8 * khalf;
    C[(size_t)m * N + col] = acc[v] + bv;
  }
}

// ---------------------------------------------------------------------------
// 3) GRU gates (PyTorch order r,z,n); writes h into hc[:,0:H] and hidden out
// ---------------------------------------------------------------------------
__global__ void gru_gates_kernel(const float* __restrict__ gi,
                                 const float* __restrict__ gh,
                                 const float* __restrict__ h_prev,
                                 float* __restrict__ hc,
                                 float* __restrict__ out_hidden) {
  int idx = blockIdx.x * blockDim.x + threadIdx.x;   // B*H
  int b = idx >> 10;
  int j = idx & 1023;
  const float* gib = gi + (size_t)b * 3 * HH;
  const float* ghb = gh + (size_t)b * 3 * HH;
  float r = 1.0f / (1.0f + expf(-(gib[j]          + ghb[j])));
  float z = 1.0f / (1.0f + expf(-(gib[HH + j]     + ghb[HH + j])));
  float n = tanhf(gib[2 * HH + j] + r * ghb[2 * HH + j]);
  float h = (1.0f - z) * n + z * h_prev[idx];
  hc[(size_t)b * 2 * HH + j] = h;
  out_hidden[idx] = h;
}

// ---------------------------------------------------------------------------
// 4) energies[s][b] = dot(h[b], enc[s][b])  — one wave per (s,b), float4 loads
// ---------------------------------------------------------------------------
__global__ void energies_kernel(const float* __restrict__ hc,
                                const float* __restrict__ enc,
                                float* __restrict__ energies) {
  int wid  = (blockIdx.x * blockDim.x + threadIdx.x) >> 5;
  int lane = threadIdx.x & 31;
  int b = wid & (BB - 1);
  int s = wid >> 6;
  const float4* e4 = (const float4*)(enc + ((size_t)s * BB + b) * HH);
  const float4* h4 = (const float4*)(hc + (size_t)b * 2 * HH);
  float p = 0.0f;
  #pragma unroll
  for (int i = 0; i < 8; ++i) {
    float4 ev = e4[i * 32 + lane];
    float4 hv = h4[i * 32 + lane];
    p += ev.x * hv.x + ev.y * hv.y + ev.z * hv.z + ev.w * hv.w;
  }
  #pragma unroll
  for (int off = 16; off > 0; off >>= 1) p += __shfl_down(p, off, 32);
  if (lane == 0) energies[s * BB + b] = p;
}

// ---------------------------------------------------------------------------
// 5) softmax over S per batch; also writes transposed attn weights (B,1,S)
// ---------------------------------------------------------------------------
__global__ void softmax_attn_kernel(const float* __restrict__ energies,
                                    float* __restrict__ weights,
                                    float* __restrict__ attn) {
  __shared__ float red[256];
  int b = blockIdx.x, tid = threadIdx.x;
  float m = -INFINITY;
  for (int s = tid; s < SS; s += 256) m = fmaxf(m, energies[s * BB + b]);
  red[tid] = m; __syncthreads();
  for (int o = 128; o > 0; o >>= 1) {
    if (tid < o) red[tid] = fmaxf(red[tid], red[tid + o]);
    __syncthreads();
  }
  m = red[0]; __syncthreads();
  float sum = 0.0f;
  for (int s = tid; s < SS; s += 256) sum += expf(energies[s * BB + b] - m);
  red[tid] = sum; __syncthreads();
  for (int o = 128; o > 0; o >>= 1) {
    if (tid < o) red[tid] += red[tid + o];
    __syncthreads();
  }
  float inv = 1.0f / red[0];
  for (int s = tid; s < SS; s += 256) {
    float w = expf(energies[s * BB + b] - m) * inv;
    weights[s * BB + b] = w;
    attn[(size_t)b * SS + s] = w;
  }
}

// ---------------------------------------------------------------------------
// 6) context[b] = sum_s w[s][b] * enc[s][b][:] ; writes hc[:,H:2H] + ctx out
// ---------------------------------------------------------------------------
__global__ void context_kernel(const float* __restrict__ weights,
                               const float* __restrict__ enc,
                               float* __restrict__ hc,
                               float* __restrict__ ctx_out) {
  int b = blockIdx.x, tid = threadIdx.x;   // 256 threads, float4 each = H
  float4 acc = make_float4(0.f, 0.f, 0.f, 0.f);
  const float4* e4 = (const float4*)enc;
  for (int s = 0; s < SS; ++s) {
    float w = weights[s * BB + b];
    float4 ev = e4[((size_t)s * BB + b) * (HH / 4) + tid];
    acc.x += w * ev.x; acc.y += w * ev.y; acc.z += w * ev.z; acc.w += w * ev.w;
  }
  ((float4*)&hc[(size_t)b * 2 * HH + HH])[tid] = acc;
  ((float4*)ctx_out)[(size_t)b * (HH / 4) + tid] = acc;
}

// ---------------------------------------------------------------------------
// 7) log_softmax over V, in place on logits
// ---------------------------------------------------------------------------
__global__ void log_softmax_kernel(float* __restrict__ out) {
  __shared__ float red[256];
  int b = blockIdx.x, tid = threadIdx.x;
  float* row = out + (size_t)b * BV;
  float m = -INFINITY;
  for (int i = tid; i < BV; i += 256) m = fmaxf(m, row[i]);
  red[tid] = m; __syncthreads();
  for (int o = 128; o > 0; o >>= 1) {
    if (tid < o) red[tid] = fmaxf(red[tid], red[tid + o]);
    __syncthreads();
  }
  m = red[0]; __syncthreads();
  float sum = 0.0f;
  for (int i = tid; i < BV; i += 256) sum += expf(row[i] - m);
  red[tid] = sum; __syncthreads();
  for (int o = 128; o > 0; o >>= 1) {
    if (tid < o) red[tid] += red[tid + o];
    __syncthreads();
  }
  float lse = m + logf(red[0]);
  for (int i = tid; i < BV; i += 256) row[i] = row[i] - lse;
}

// ---------------------------------------------------------------------------
extern "C" void kernel_launch(void* const* d_in, const int* in_sizes, int n_in,
                              void* d_out, int out_size, void* d_ws, size_t ws_size,
                              hipStream_t stream) {
  const int*   word_input   = (const int*)  d_in[0];
  const float* last_context = (const float*)d_in[1];
  const float* last_hidden  = (const float*)d_in[2];
  const float* enc          = (const float*)d_in[3];
  const float* emb_table    = (const float*)d_in[4];
  const float* w_ih         = (const float*)d_in[5];
  const float* w_hh         = (const float*)d_in[6];
  const float* b_ih         = (const float*)d_in[7];
  const float* b_hh         = (const float*)d_in[8];
  const float* out_W        = (const float*)d_in[9];
  const float* out_b        = (const float*)d_in[10];

  float* out         = (float*)d_out;
  float* out_output  = out;                            // (B, V)
  float* out_context = out_output + (size_t)BB * BV;   // (B, H)
  float* out_hidden  = out_context + (size_t)BB * HH;  // (1, B, H)
  float* out_attn    = out_hidden + (size_t)BB * HH;   // (B, 1, S)

  float* ws       = (float*)d_ws;
  float* x0       = ws;                      // 64*2048
  float* gi       = x0 + BB * (EE + HH);     // 64*3072
  float* gh       = gi + BB * 3 * HH;        // 64*3072
  float* hc       = gh + BB * 3 * HH;        // 64*2048 (h | context)
  float* energies = hc + BB * 2 * HH;        // 1024*64
  float* weights  = energies + SS * BB;      // 1024*64

  // 1) x0
  build_x0_kernel<<<(BB * (EE + HH)) / 256, 256, 0, stream>>>(
      word_input, last_context, emb_table, x0);

  // 2) gi = x0 @ w_ih^T + b_ih   (N=3072, K=2048)
  wmma_gemm_bias_kernel<<<3 * HH / 64, 512, 0, stream>>>(
      x0, w_ih, b_ih, gi, 3 * HH, EE + HH);
  //    gh = h_prev @ w_hh^T + b_hh (N=3072, K=1024)
  wmma_gemm_bias_kernel<<<3 * HH / 64, 512, 0, stream>>>(
      last_hidden, w_hh, b_hh, gh, 3 * HH, HH);

  // 3) GRU gates -> h (into hc[:,0:H] and hidden output)
  gru_gates_kernel<<<(BB * HH) / 256, 256, 0, stream>>>(
      gi, gh, last_hidden, hc, out_hidden);

  // 4) energies (S,B) — one wave per (s,b)
  energies_kernel<<<(SS * BB) / 8, 256, 0, stream>>>(hc, enc, energies);

  // 5) softmax over S + attn weights output
  softmax_attn_kernel<<<BB, 256, 0, stream>>>(energies, weights, out_attn);

  // 6) context (into hc[:,H:2H] and context output)
  context_kernel<<<BB, 256, 0, stream>>>(weights, enc, hc, out_context);

  // 7) logits = hc @ out_W^T + out_b  (N=32000, K=2048) -> d_out
  wmma_gemm_bias_kernel<<<BV / 64, 512, 0, stream>>>(
      hc, out_W, out_b, out_output, BV, EE + HH);

  // 8) log_softmax in place
  log_softmax_kernel<<<BB, 256, 0, stream>>>(out_output);
}